// Model_89163521065606
// MI455X (gfx1250) — compile-verified
//
#include <hip/hip_runtime.h>
#include <hip/hip_bf16.h>
#include <math.h>

// ---------------- problem constants ----------------
#define BSZ   64
#define TT    10
#define XX    256
#define GRD   (TT * XX)          // 2560
#define LAMv  100.0f
#define CGIT  200
#define NTH   256
#define CC    64                 // conv ci-chunk staged in LDS

// ---------------- workspace layout (float offsets) ----------------
#define WS_PARAMS 0              // 4 floats (MLP output z0..z3)
#define WS_CT1    16             // 200 floats: ct1, ct2, ct1T, ct2T (each 10*5)
#define WS_CX1    256            // 5120 floats: cx1, cx2, cx1T, cx2T (each 256*5)
#define WS_CONVA  8192           // 64*256*256 floats
#define WS_CONVB  (WS_CONVA + 64 * 256 * 256)

typedef float v2f __attribute__((ext_vector_type(2)));
typedef float v8f __attribute__((ext_vector_type(8)));

// =====================================================================
// MLP: param_in(1,64) -> 1024 -> 1024 -> 1024 -> 4, ELU between layers.
// One block of 256 threads; everything in LDS.
// =====================================================================
__global__ __launch_bounds__(256) void mlp_kernel(
    const float* __restrict__ pin,
    const float* __restrict__ w0, const float* __restrict__ b0,
    const float* __restrict__ w1, const float* __restrict__ b1,
    const float* __restrict__ w2, const float* __restrict__ b2,
    const float* __restrict__ w3, const float* __restrict__ b3,
    float* __restrict__ params_out)
{
    __shared__ float za[1024];
    __shared__ float zb[1024];
    const int tid = threadIdx.x;

    if (tid < 64) za[tid] = pin[tid];
    __syncthreads();

    // L0: 64 -> 1024, ELU
    for (int f = tid; f < 1024; f += NTH) {
        float s = b0[f];
        const float* wr = w0 + (size_t)f * 64;
        for (int i = 0; i < 64; ++i) s = fmaf(za[i], wr[i], s);
        zb[f] = (s > 0.f) ? s : (expf(s) - 1.f);
    }
    __syncthreads();
    // L1: 1024 -> 1024, ELU
    for (int f = tid; f < 1024; f += NTH) {
        float s = b1[f];
        const float* wr = w1 + (size_t)f * 1024;
        for (int i = 0; i < 1024; ++i) s = fmaf(zb[i], wr[i], s);
        za[f] = (s > 0.f) ? s : (expf(s) - 1.f);
    }
    __syncthreads();
    // L2: 1024 -> 1024, ELU
    for (int f = tid; f < 1024; f += NTH) {
        float s = b2[f];
        const float* wr = w2 + (size_t)f * 1024;
        for (int i = 0; i < 1024; ++i) s = fmaf(za[i], wr[i], s);
        zb[f] = (s > 0.f) ? s : (expf(s) - 1.f);
    }
    __syncthreads();
    // L3: 1024 -> 4, linear
    if (tid < 4) {
        float s = b3[tid];
        const float* wr = w3 + (size_t)tid * 1024;
        for (int i = 0; i < 1024; ++i) s = fmaf(zb[i], wr[i], s);
        params_out[tid] = s;
    }
}

// =====================================================================
// p,q elementwise: p = z0 + z1*u ; q = z2 + z3*u
// =====================================================================
__global__ void pq_kernel(const float* __restrict__ u,
                          const float* __restrict__ params,
                          float* __restrict__ outp, float* __restrict__ outq)
{
    int i = blockIdx.x * blockDim.x + threadIdx.x;
    if (i < BSZ * GRD) {
        float uu = u[i];
        outp[i] = fmaf(params[1], uu, params[0]);
        outq[i] = fmaf(params[3], uu, params[2]);
    }
}

// =====================================================================
// Finite-difference operators as centered width-5 bands:
//   band[j][o+2] == M[j][j+o], o in [-2,2]  (covers D2's copied edge rows)
// plus transposed bands: bandT[k][o+2] = M[k+o][k] = band[k+o][2-o].
// =====================================================================
__device__ inline void fill_bands(int j, int n, const float* h,
                                  float* c1row, float* c2row)
{
    #pragma unroll
    for (int i = 0; i < 5; ++i) { c1row[i] = 0.f; c2row[i] = 0.f; }
    // D1
    if (j == 0) {
        c1row[2] = -1.f / h[0];     c1row[3] = 1.f / h[0];
    } else if (j == n - 1) {
        c1row[1] = -1.f / h[n - 2]; c1row[2] = 1.f / h[n - 2];
    } else {
        float hm = h[j - 1], hp = h[j];
        c1row[1] = -hp / (hm * (hm + hp));
        c1row[2] = (hp - hm) / (hm * hp);
        c1row[3] = hm / (hp * (hm + hp));
    }
    // D2 (rows 0 / n-1 copy rows 1 / n-2 at those rows' columns)
    if (j == 0) {
        float hm = h[0], hp = h[1];
        c2row[2] = 2.f / (hm * (hm + hp));
        c2row[3] = -2.f / (hm * hp);
        c2row[4] = 2.f / (hp * (hm + hp));
    } else if (j == n - 1) {
        float hm = h[n - 3], hp = h[n - 2];
        c2row[0] = 2.f / (hm * (hm + hp));
        c2row[1] = -2.f / (hm * hp);
        c2row[2] = 2.f / (hp * (hm + hp));
    } else {
        float hm = h[j - 1], hp = h[j];
        c2row[1] = 2.f / (hm * (hm + hp));
        c2row[2] = -2.f / (hm * hp);
        c2row[3] = 2.f / (hp * (hm + hp));
    }
}

__global__ __launch_bounds__(256) void dmat_kernel(
    const float* __restrict__ steps0, const float* __restrict__ steps1,
    float* __restrict__ ws)
{
    __shared__ float ht[TT - 1];
    __shared__ float hx[XX - 1];
    __shared__ float bt[200];    // ct1(50) ct2(50) ct1T(50) ct2T(50)
    __shared__ float bx[5120];   // cx1(1280) cx2(1280) cx1T(1280) cx2T(1280)
    const int tid = threadIdx.x;

    if (tid < TT - 1) {
        float s = 1.f / (1.f + expf(-steps0[tid]));
        ht[tid] = fminf(fmaxf(s, 0.005f), 0.1f);
    }
    if (tid < XX - 1) {
        float s = 1.f / (1.f + expf(-steps1[tid]));
        hx[tid] = fminf(fmaxf(s, 0.005f), 0.1f);
    }
    __syncthreads();

    if (tid < TT)  fill_bands(tid, TT, ht, &bt[tid * 5], &bt[50 + tid * 5]);
    if (tid < XX)  fill_bands(tid, XX, hx, &bx[tid * 5], &bx[1280 + tid * 5]);
    __syncthreads();

    if (tid < TT) {
        #pragma unroll
        for (int o = -2; o <= 2; ++o) {
            int s = tid + o;
            bt[100 + tid * 5 + o + 2] = ((unsigned)s < TT) ? bt[s * 5 + (2 - o)] : 0.f;
            bt[150 + tid * 5 + o + 2] = ((unsigned)s < TT) ? bt[50 + s * 5 + (2 - o)] : 0.f;
        }
    }
    if (tid < XX) {
        #pragma unroll
        for (int o = -2; o <= 2; ++o) {
            int s = tid + o;
            bx[2560 + tid * 5 + o + 2] = ((unsigned)s < XX) ? bx[s * 5 + (2 - o)] : 0.f;
            bx[3840 + tid * 5 + o + 2] = ((unsigned)s < XX) ? bx[1280 + s * 5 + (2 - o)] : 0.f;
        }
    }
    __syncthreads();

    for (int i = tid; i < 200; i += NTH)  ws[WS_CT1 + i] = bt[i];
    for (int i = tid; i < 5120; i += NTH) ws[WS_CX1 + i] = bx[i];
}

// =====================================================================
// Conv1d (k=5, pad 2) as LDS-staged implicit GEMM on V_WMMA_F32_16X16X4_F32.
//   Block: 128 threads = 4 wave32, covering 64 x-positions (16 per wave)
//   and 16 output channels. K = cin*5, processed in ci-chunks of 64
//   (K-chunk 320) staged in LDS, zero-padded so the WMMA loop is guard-free.
//   A[m][k] = w[co0+m][cib*5+k]   (LDS, row stride 325 -> conflict-free)
//   B[k][nn] = in[n][cib+k/5][xb + wv*16 + nn + k%5 - 2]  (LDS, halo'd)
//   K unrolled by 16: 4 v_wmma_f32_16x16x4_f32 per body, chained on one accumulator.
// =====================================================================
__global__ __launch_bounds__(128) void conv_wmma_kernel(
    const float* __restrict__ in, const float* __restrict__ w,
    const float* __restrict__ bias, float* __restrict__ out,
    int cin, int cout, int relu)
{
    __shared__ float w_s[16 * 325];   // 16 cout rows x up to 320 K (pad 325)
    __shared__ float in_s[CC * 68];   // 64 ci rows x (64 + 4 halo) x-positions

    const int tid  = threadIdx.x;
    const int lane = tid & 31;
    const int wv   = tid >> 5;
    const int n    = blockIdx.x;
    const int co0  = blockIdx.y * 16;
    const int xb   = blockIdx.z * 64;
    const int row  = lane & 15;
    const int hi   = lane >> 4;
    const int K    = cin * 5;

    const float* inb = in + (size_t)n * cin * 256;

    v8f acc = {0.f, 0.f, 0.f, 0.f, 0.f, 0.f, 0.f, 0.f};

    for (int cib = 0; cib < cin; cib += CC) {
        const int cc  = (cin - cib < CC) ? (cin - cib) : CC;
        const int KC  = cc * 5;
        const int KCp = (KC + 15) & ~15;

        __syncthreads();   // previous chunk's WMMA reads done before restage

        // stage weight tile (zero-pad K to KCp and masked cout rows)
        for (int m = 0; m < 16; ++m) {
            const int co = co0 + m;
            const float* wr = w + (size_t)co * K + (size_t)cib * 5;
            for (int k = tid; k < KCp; k += 128)
                w_s[m * 325 + k] = (co < cout && k < KC) ? wr[k] : 0.f;
        }
        // stage input tile with halo; rows >= cc and OOB x zero-filled
        for (int idx = tid; idx < CC * 68; idx += 128) {
            const int cil = idx / 68;
            const int xl  = idx - cil * 68;
            const int xi  = xb + xl - 2;
            float v = 0.f;
            if (cil < cc && xi >= 0 && xi < 256)
                v = inb[(size_t)(cib + cil) * 256 + xi];
            in_s[idx] = v;
        }
        // pull next weight chunk toward GL2 while this chunk computes
        if (cib + CC < cin)
            __builtin_prefetch(w + (size_t)co0 * K + (size_t)(cib + CC) * 5, 0, 1);
        __syncthreads();

        auto step = [&](int kb) {
            const int k0  = kb + hi * 2;
            const int k1  = k0 + 1;
            const int ci0 = k0 / 5, kk0 = k0 - ci0 * 5;
            const int ci1 = k1 / 5, kk1 = k1 - ci1 * 5;
            v2f a, b;
            a.x = w_s[row * 325 + k0];
            a.y = w_s[row * 325 + k1];
            b.x = in_s[ci0 * 68 + wv * 16 + row + kk0];
            b.y = in_s[ci1 * 68 + wv * 16 + row + kk1];
            acc = __builtin_amdgcn_wmma_f32_16x16x4_f32(
                /*neg_a=*/false, a, /*neg_b=*/false, b,
                /*c_mod=*/(short)0, acc, /*reuse_a=*/false, /*reuse_b=*/false);
        };
        for (int kb = 0; kb < KCp; kb += 16) {
            step(kb); step(kb + 4); step(kb + 8); step(kb + 12);
        }
    }

    const int x = xb + wv * 16 + row;
    #pragma unroll
    for (int r = 0; r < 8; ++r) {
        const int co = co0 + r + hi * 8;
        if (co < cout) {
            float v = acc[r] + bias[co];
            if (relu) v = fmaxf(v, 0.f);
            out[((size_t)n * cout + co) * 256 + x] = v;
        }
    }
}

// =====================================================================
// CG solve per batch sample, banded-stencil operators, all state in LDS.
//   A(v)  = D1t v + p .* (v D1x^T) + q .* (v D2x^T)
//   gram  = A^T A + LAM * boundary_mask
//   b     = LAM * boundary_mask .* u     (B^T of boundary values of u)
// One block (8 wave32) per sample; 200 fixed iterations.
// =====================================================================
__global__ __launch_bounds__(256) void cg_kernel(
    const float* __restrict__ u, const float* __restrict__ params,
    const float* __restrict__ ws,
    float* __restrict__ out_u0, float* __restrict__ out_eps)
{
    extern __shared__ float sm[];
    float* xs   = sm;               // 2560 solution
    float* rr   = xs + GRD;         // residual
    float* pv   = rr + GRD;         // search direction
    float* Ap   = pv + GRD;         // gram(pv)
    float* w1   = Ap + GRD;         // temp: A(v)
    float* us   = w1 + GRD;         // u for this sample
    float* bt1  = us + GRD;         // 50
    float* bt2  = bt1 + 50;
    float* bt1T = bt2 + 50;
    float* bt2T = bt1T + 50;
    float* bx1  = bt2T + 50;        // 1280
    float* bx2  = bx1 + 1280;
    float* bx1T = bx2 + 1280;
    float* bx2T = bx1T + 1280;
    float* red  = bx2T + 1280;      // 256 reduction scratch

    const int tid = threadIdx.x;
    const int n   = blockIdx.x;

    const float pz0 = params[0], pz1 = params[1];
    const float qz0 = params[2], qz1 = params[3];

    for (int i = tid; i < GRD; i += NTH) us[i] = u[(size_t)n * GRD + i];
    for (int i = tid; i < 200; i += NTH)  bt1[i] = ws[WS_CT1 + i];
    for (int i = tid; i < 5120; i += NTH) bx1[i] = ws[WS_CX1 + i];
    __syncthreads();

    // ---- helpers ----
    auto applyA = [&](const float* src, float* dst) {
        for (int i = tid; i < GRD; i += NTH) {
            int t = i >> 8, j = i & 255;
            float ut = 0.f;
            #pragma unroll
            for (int o = -2; o <= 2; ++o) {
                int tt = t + o;
                if ((unsigned)tt < TT) ut = fmaf(bt1[t * 5 + o + 2], src[tt * XX + j], ut);
            }
            float ux = 0.f, uxx = 0.f;
            #pragma unroll
            for (int o = -2; o <= 2; ++o) {
                int jj = j + o;
                if ((unsigned)jj < XX) {
                    float v = src[t * XX + jj];
                    ux  = fmaf(bx1[j * 5 + o + 2], v, ux);
                    uxx = fmaf(bx2[j * 5 + o + 2], v, uxx);
                }
            }
            float uu = us[i];
            dst[i] = ut + fmaf(pz1, uu, pz0) * ux + fmaf(qz1, uu, qz0) * uxx;
        }
    };
    // dst = A^T(src) + LAM * mask .* reg
    auto applyATreg = [&](const float* src, const float* reg, float* dst) {
        for (int i = tid; i < GRD; i += NTH) {
            int t = i >> 8, k = i & 255;
            float a = 0.f;
            #pragma unroll
            for (int o = -2; o <= 2; ++o) {
                int tt = t + o;
                if ((unsigned)tt < TT) a = fmaf(bt1T[t * 5 + o + 2], src[tt * XX + k], a);
            }
            #pragma unroll
            for (int o = -2; o <= 2; ++o) {
                int kk = k + o;
                if ((unsigned)kk < XX) {
                    float v  = src[t * XX + kk];
                    float uu = us[t * XX + kk];
                    a = fmaf(bx1T[k * 5 + o + 2], fmaf(pz1, uu, pz0) * v, a);
                    a = fmaf(bx2T[k * 5 + o + 2], fmaf(qz1, uu, qz0) * v, a);
                }
            }
            float m = (t == 0 || t == TT - 1 || k == 0 || k == XX - 1) ? LAMv : 0.f;
            dst[i] = fmaf(m, reg[i], a);
        }
    };
    auto blockDot = [&](const float* a, const float* b) -> float {
        float s = 0.f;
        for (int i = tid; i < GRD; i += NTH) s = fmaf(a[i], b[i], s);
        red[tid] = s;
        __syncthreads();
        for (int off = NTH / 2; off > 0; off >>= 1) {
            if (tid < off) red[tid] += red[tid + off];
            __syncthreads();
        }
        float r = red[0];
        __syncthreads();
        return r;
    };

    // ---- init: x=0, r=p=b=LAM*mask.*u ----
    for (int i = tid; i < GRD; i += NTH) {
        int t = i >> 8, j = i & 255;
        float m = (t == 0 || t == TT - 1 || j == 0 || j == XX - 1) ? LAMv : 0.f;
        float b = m * us[i];
        xs[i] = 0.f; rr[i] = b; pv[i] = b;
    }
    __syncthreads();

    float rs = blockDot(rr, rr);

    for (int it = 0; it < CGIT; ++it) {
        applyA(pv, w1);
        __syncthreads();
        applyATreg(w1, pv, Ap);
        __syncthreads();
        float pAp   = blockDot(pv, Ap);
        float alpha = rs / fmaxf(pAp, 1e-30f);
        for (int i = tid; i < GRD; i += NTH) {
            xs[i] = fmaf(alpha, pv[i], xs[i]);
            rr[i] = fmaf(-alpha, Ap[i], rr[i]);
        }
        __syncthreads();
        float rs2  = blockDot(rr, rr);
        float beta = rs2 / fmaxf(rs, 1e-30f);
        for (int i = tid; i < GRD; i += NTH)
            pv[i] = fmaf(beta, pv[i], rr[i]);
        __syncthreads();
        rs = rs2;
    }

    // eps = ||A(x)||  (rhs r is zero in the reference)
    applyA(xs, w1);
    __syncthreads();
    float s = 0.f;
    for (int i = tid; i < GRD; i += NTH) s = fmaf(w1[i], w1[i], s);
    red[tid] = s;
    __syncthreads();
    for (int off = NTH / 2; off > 0; off >>= 1) {
        if (tid < off) red[tid] += red[tid + off];
        __syncthreads();
    }
    if (tid == 0) out_eps[n] = sqrtf(red[0]);
    for (int i = tid; i < GRD; i += NTH) out_u0[(size_t)n * GRD + i] = xs[i];
}

// =====================================================================
// Host-side orchestration
// Input order: 0:u 1:t 2:x 3..9:conv_w 10..16:conv_b 17..20:mlp_w
//              21..24:mlp_b 25:param_in 26:steps0 27:steps1
// Output: u0[64*2560] | up[64*2560] | eps[64] | p[64*2560] | q[64*2560]
// =====================================================================
extern "C" void kernel_launch(void* const* d_in, const int* in_sizes, int n_in,
                              void* d_out, int out_size, void* d_ws, size_t ws_size,
                              hipStream_t stream)
{
    (void)in_sizes; (void)n_in; (void)out_size; (void)ws_size;

    const float* u = (const float*)d_in[0];
    const float* conv_w[7]; const float* conv_b[7];
    for (int i = 0; i < 7; ++i) {
        conv_w[i] = (const float*)d_in[3 + i];
        conv_b[i] = (const float*)d_in[10 + i];
    }
    const float* mlp_w[4]; const float* mlp_b[4];
    for (int i = 0; i < 4; ++i) {
        mlp_w[i] = (const float*)d_in[17 + i];
        mlp_b[i] = (const float*)d_in[21 + i];
    }
    const float* param_in = (const float*)d_in[25];
    const float* steps0   = (const float*)d_in[26];
    const float* steps1   = (const float*)d_in[27];

    float* ws  = (float*)d_ws;
    float* out = (float*)d_out;
    float* out_u0  = out;
    float* out_up  = out + BSZ * GRD;
    float* out_eps = out + 2 * BSZ * GRD;
    float* out_p   = out_eps + BSZ;
    float* out_q   = out_p + BSZ * GRD;

    // MLP -> 4 scalar params
    mlp_kernel<<<1, 256, 0, stream>>>(param_in,
        mlp_w[0], mlp_b[0], mlp_w[1], mlp_b[1],
        mlp_w[2], mlp_b[2], mlp_w[3], mlp_b[3], ws + WS_PARAMS);

    // p, q outputs
    pq_kernel<<<(BSZ * GRD + 255) / 256, 256, 0, stream>>>(
        u, ws + WS_PARAMS, out_p, out_q);

    // FD operator bands
    dmat_kernel<<<1, 256, 0, stream>>>(steps0, steps1, ws);

    // conv stack (LDS-staged WMMA implicit GEMM), ping-pong through workspace
    static const int CH[8] = {10, 64, 128, 256, 256, 128, 64, 10};
    const float* cur = u;
    for (int i = 0; i < 7; ++i) {
        int cin = CH[i], cout = CH[i + 1];
        float* o = (i == 6) ? out_up
                            : (ws + ((i & 1) ? WS_CONVB : WS_CONVA));
        dim3 grid(BSZ, (cout + 15) / 16, 4);
        conv_wmma_kernel<<<grid, 128, 0, stream>>>(
            cur, conv_w[i], conv_b[i], o, cin, cout, (i < 6) ? 1 : 0);
        cur = o;
    }

    // CG solves: 64 samples, one workgroup each, LDS-resident
    const size_t smemBytes =
        (size_t)(6 * GRD + 200 + 5120 + NTH) * sizeof(float); // ~83.7 KB
    cg_kernel<<<BSZ, NTH, smemBytes, stream>>>(
        u, ws + WS_PARAMS, ws, out_u0, out_eps);
}